// LightGCN_14817637171647
// MI455X (gfx1250) — compile-verified
//
#include <hip/hip_runtime.h>
#include <stdint.h>

#define NUM_USERS 100000
#define NUM_ITEMS 50000
#define NUM_NODES (NUM_USERS + NUM_ITEMS)
#define EMB_DIM   64
#define TILE_EDGES 256
#define BLOCK      256

// -------- wait helper: CDNA5 ASYNCcnt --------
__device__ __forceinline__ void wait_asynccnt0() {
#if __has_builtin(__builtin_amdgcn_s_wait_asynccnt)
    __builtin_amdgcn_s_wait_asynccnt(0);
#else
    asm volatile("s_wait_asynccnt 0" ::: "memory");
#endif
}

// -------- async global -> LDS 4-byte copy (CDNA5) --------
// vdst = LDS byte address (low 32 bits of flat shared pointer), vaddr = 64-bit global VA
__device__ __forceinline__ void async_copy_b32(const void* gaddr, void* lds_ptr) {
    unsigned lds_addr = (unsigned)(size_t)lds_ptr;            // addrspace(3) offset lives in low 32 bits
    unsigned long long ga = (unsigned long long)(size_t)gaddr;
    asm volatile("global_load_async_to_lds_b32 %0, %1, off"
                 :: "v"(lds_addr), "v"(ga)
                 : "memory");
}

// y += A * x  (COO scatter SpMM). 16 lanes per edge, float4 per lane (256B/edge).
// Edge metadata staged per 256-edge tile into LDS via CDNA5 async global->LDS copies.
// FIRST kernel in the file so the disasm snippet shows its codegen.
__global__ void lgcn_spmm_kernel(const int*   __restrict__ rows,
                                 const int*   __restrict__ cols,
                                 const float* __restrict__ vals,
                                 const float* __restrict__ x,
                                 float*       __restrict__ y,
                                 int nEdges) {
    __shared__ int   s_rows[TILE_EDGES];
    __shared__ int   s_cols[TILE_EDGES];
    __shared__ float s_vals[TILE_EDGES];

    const int tid  = threadIdx.x;
    const long long base = (long long)blockIdx.x * TILE_EDGES;
    const long long e = base + tid;

    // ---- stage this block's edge tile into LDS (ASYNCcnt-tracked copies) ----
    if (e < (long long)nEdges) {
        async_copy_b32(rows + e, &s_rows[tid]);
        async_copy_b32(cols + e, &s_cols[tid]);
        async_copy_b32(vals + e, &s_vals[tid]);
    }
    wait_asynccnt0();
    __syncthreads();

    const int lane = tid & 15;   // 16 lanes x float4 = 64 floats = one edge row
    const int grp  = tid >> 4;   // 16 edge-groups per block

    #pragma unroll 4
    for (int j = 0; j < TILE_EDGES / 16; ++j) {
        const int s = j * 16 + grp;
        if (base + s < (long long)nEdges) {
            const int   r = s_rows[s];   // LDS broadcast across the 16-lane group
            const int   c = s_cols[s];
            const float v = s_vals[s];

            const float4* xr = (const float4*)(x + (size_t)c * EMB_DIM);
            float4 xv = xr[lane];                       // global_load_b128, coalesced 256B

            float* yr = y + (size_t)r * EMB_DIM + (size_t)lane * 4;
            atomicAdd(yr + 0, xv.x * v);                // expect global_atomic_add_f32 (no return)
            atomicAdd(yr + 1, xv.y * v);
            atomicAdd(yr + 2, xv.z * v);
            atomicAdd(yr + 3, xv.w * v);
        }
    }
}

// out = emb ; zbuf = 0        (n4 float4 elements)
__global__ void lgcn_init_kernel(const float4* __restrict__ emb,
                                 float4* __restrict__ out,
                                 float4* __restrict__ zbuf, int n4) {
    int i = blockIdx.x * blockDim.x + threadIdx.x;
    if (i < n4) {
        out[i]  = emb[i];
        zbuf[i] = make_float4(0.f, 0.f, 0.f, 0.f);
    }
}

// out += src*coef ; optionally zbuf = 0
__global__ void lgcn_axpy_zero_kernel(const float4* __restrict__ src,
                                      float4* __restrict__ out,
                                      float4* __restrict__ zbuf,
                                      float coef, int n4, int do_zero) {
    int i = blockIdx.x * blockDim.x + threadIdx.x;
    if (i < n4) {
        float4 s = src[i];
        float4 o = out[i];
        o.x += s.x * coef; o.y += s.y * coef; o.z += s.z * coef; o.w += s.w * coef;
        out[i] = o;
        if (do_zero) zbuf[i] = make_float4(0.f, 0.f, 0.f, 0.f);
    }
}

extern "C" void kernel_launch(void* const* d_in, const int* in_sizes, int n_in,
                              void* d_out, int out_size, void* d_ws, size_t ws_size,
                              hipStream_t stream) {
    const int*   rows = (const int*)d_in[0];
    const int*   cols = (const int*)d_in[1];
    const float* vals = (const float*)d_in[2];
    const float* emb  = (const float*)d_in[3];
    float*       out  = (float*)d_out;
    const int nE = in_sizes[0];

    const long long nFeat = (long long)NUM_NODES * EMB_DIM;   // 9.6M floats
    float* buf0 = (float*)d_ws;                                // needs 2 * 38.4MB of ws
    float* buf1 = buf0 + nFeat;

    const int n4 = (int)(nFeat / 4);
    dim3 blk(BLOCK);
    dim3 gFeat((n4 + BLOCK - 1) / BLOCK);
    dim3 gTile((nE + TILE_EDGES - 1) / TILE_EDGES);

    // out = emb ; buf0 = 0
    lgcn_init_kernel<<<gFeat, blk, 0, stream>>>((const float4*)emb, (float4*)out,
                                                (float4*)buf0, n4);
    // layer 1: buf0 += A*emb
    lgcn_spmm_kernel<<<gTile, blk, 0, stream>>>(rows, cols, vals, emb, buf0, nE);
    // out += buf0/2 ; buf1 = 0
    lgcn_axpy_zero_kernel<<<gFeat, blk, 0, stream>>>((const float4*)buf0, (float4*)out,
                                                     (float4*)buf1, 0.5f, n4, 1);
    // layer 2: buf1 += A*buf0
    lgcn_spmm_kernel<<<gTile, blk, 0, stream>>>(rows, cols, vals, buf0, buf1, nE);
    // out += buf1/3 ; buf0 = 0
    lgcn_axpy_zero_kernel<<<gFeat, blk, 0, stream>>>((const float4*)buf1, (float4*)out,
                                                     (float4*)buf0, 1.0f / 3.0f, n4, 1);
    // layer 3: buf0 += A*buf1
    lgcn_spmm_kernel<<<gTile, blk, 0, stream>>>(rows, cols, vals, buf1, buf0, nE);
    // out += buf0/4
    lgcn_axpy_zero_kernel<<<gFeat, blk, 0, stream>>>((const float4*)buf0, (float4*)out,
                                                     (float4*)buf0, 0.25f, n4, 0);
}